// GraphFormerDecoder_84284438217361
// MI455X (gfx1250) — compile-verified
//
#include <hip/hip_runtime.h>

// Problem constants (match reference)
#define NNODES   65536
#define NEDGES   1048576
#define NB       4
#define NODE_PER 16384
#define DIN      64
#define HIDC     32
#define EMBC     32
#define NHEAD    4
#define NCLS     2
#define HC       128          // NHEAD * 32 (q/k/v row width, all layers)
#define INV_SQRT_C 0.17677669529663687f   // 1/sqrt(32)

typedef __attribute__((ext_vector_type(2))) float v2f;
typedef __attribute__((ext_vector_type(8))) float v8f;

// ---------------------------------------------------------------------------
// FP32 WMMA GEMM:  Y[N x Ncols] = X[N x K] @ W[K x Ncols] + bias
// One wave32 per 16x16 output tile. EXEC is all-ones by construction
// (grid covers tiles exactly), as required for V_WMMA.
//
// A 16x4 f32 layout: lanes 0-15 -> M=lane, {v0,v1}={K=0,1}; lanes 16-31 ->
// M=lane-16, {v0,v1}={K=2,3}. B mirrors A with N in place of M.
// C/D 16x16 f32: acc[r] -> row r (lanes 0-15) / row r+8 (lanes 16-31),
// col = lane & 15.
// ---------------------------------------------------------------------------
__global__ void gemm_bias_wmma(const float* __restrict__ X,
                               const float* __restrict__ W,
                               const float* __restrict__ bias,
                               float* __restrict__ Y,
                               int K, int Ncols) {
  const int lane   = threadIdx.x & 31;
  const int wave   = blockIdx.x * (blockDim.x >> 5) + (threadIdx.x >> 5);
  const int tilesN = Ncols >> 4;
  const int tileM  = wave / tilesN;
  const int tileN  = wave % tilesN;
  const int l15    = lane & 15;
  const int khalf  = lane >> 4;            // 0 or 1
  const int rowBase = tileM << 4;
  const int colBase = tileN << 4;

  const float* xrow = X + (size_t)(rowBase + l15) * K;
  const float* wcol = W + colBase + l15;

  v8f acc = {};
  for (int k0 = 0; k0 < K; k0 += 4) {
    const int ka = k0 + khalf * 2;
    v2f a, b;
    a.x = xrow[ka];
    a.y = xrow[ka + 1];
    b.x = wcol[(size_t)ka * Ncols];
    b.y = wcol[(size_t)(ka + 1) * Ncols];
    // D = A*B + C  (fp32, 16x16x4)
    acc = __builtin_amdgcn_wmma_f32_16x16x4_f32(false, a, false, b,
                                                (short)0, acc, false, false);
  }

  const float bn   = bias[colBase + l15];
  float* ycol      = Y + colBase + l15;
  const int rOff   = rowBase + khalf * 8;
#pragma unroll
  for (int r = 0; r < 8; ++r) {
    ycol[(size_t)(rOff + r) * Ncols] = acc[r] + bn;
  }
}

// ---------------------------------------------------------------------------
// Monotonic key mapping: float total order -> unsigned total order
// ---------------------------------------------------------------------------
__device__ __forceinline__ unsigned fkey(float f) {
  unsigned u = __float_as_uint(f);
  return (u & 0x80000000u) ? ~u : (u | 0x80000000u);
}
__device__ __forceinline__ float funkey(unsigned k) {
  unsigned u = (k & 0x80000000u) ? (k & 0x7fffffffu) : ~k;
  return __uint_as_float(u);
}

__global__ void init_seg(unsigned* __restrict__ mkey, float* __restrict__ ssum, int n) {
  int i = blockIdx.x * blockDim.x + threadIdx.x;
  if (i < n) {
    mkey[i] = fkey(-__builtin_inff());
    ssum[i] = 0.0f;
  }
}

__global__ void zero_f32(float* __restrict__ p, int n) {
  int i = blockIdx.x * blockDim.x + threadIdx.x;
  if (i < n) p[i] = 0.0f;
}

// alpha[e,h] = dot(q[dst[e],h,:], k[src[e],h,:]) / sqrt(C)
__global__ void edge_alpha(const float* __restrict__ q, const float* __restrict__ k,
                           const int* __restrict__ src, const int* __restrict__ dst,
                           float* __restrict__ alpha) {
  int idx = blockIdx.x * blockDim.x + threadIdx.x;   // e*NHEAD + h
  if (idx >= NEDGES * NHEAD) return;
  int e = idx >> 2;
  int h = idx & 3;
  if (e + 2048 < NEDGES) {                 // pull index cachelines ahead
    __builtin_prefetch(src + e + 2048, 0, 0);
    __builtin_prefetch(dst + e + 2048, 0, 0);
  }
  const float* qr = q + (size_t)dst[e] * HC + h * 32;
  const float* kr = k + (size_t)src[e] * HC + h * 32;
  float s = 0.0f;
#pragma unroll
  for (int c = 0; c < 32; c += 4) {
    float4 qv = *(const float4*)(qr + c);
    float4 kv = *(const float4*)(kr + c);
    s += qv.x * kv.x + qv.y * kv.y + qv.z * kv.z + qv.w * kv.w;
  }
  alpha[idx] = s * INV_SQRT_C;
}

// segment max over destination neighborhoods
__global__ void edge_max(const float* __restrict__ alpha, const int* __restrict__ dst,
                         unsigned* __restrict__ mkey) {
  int idx = blockIdx.x * blockDim.x + threadIdx.x;
  if (idx >= NEDGES * NHEAD) return;
  int e = idx >> 2;
  int h = idx & 3;
  atomicMax(&mkey[(size_t)dst[e] * NHEAD + h], fkey(alpha[idx]));
}

// e = exp(alpha - m[dst]); alpha <- e ; ssum[dst] += e
__global__ void edge_expsum(float* __restrict__ alpha, const int* __restrict__ dst,
                            const unsigned* __restrict__ mkey, float* __restrict__ ssum) {
  int idx = blockIdx.x * blockDim.x + threadIdx.x;
  if (idx >= NEDGES * NHEAD) return;
  int e = idx >> 2;
  int h = idx & 3;
  size_t seg = (size_t)dst[e] * NHEAD + h;
  float m = funkey(mkey[seg]);           // finite for any segment with edges
  float ev = expf(alpha[idx] - m);
  alpha[idx] = ev;
  atomicAdd(&ssum[seg], ev);
}

// alpha <- alpha / (ssum[dst] + eps)   (finished attention coefficient)
__global__ void edge_norm(float* __restrict__ alpha, const int* __restrict__ dst,
                          const float* __restrict__ ssum) {
  int idx = blockIdx.x * blockDim.x + threadIdx.x;
  if (idx >= NEDGES * NHEAD) return;
  int e = idx >> 2;
  int h = idx & 3;
  alpha[idx] = alpha[idx] / (ssum[(size_t)dst[e] * NHEAD + h] + 1e-16f);
}

// One wave32 per edge; lane c owns channel c for all heads.
// Each wave issues 4 fully-coalesced 128B atomic bursts into agg[dst].
__global__ void edge_scatter(const float* __restrict__ alpha, const int* __restrict__ src,
                             const int* __restrict__ dst, const float* __restrict__ v,
                             float* __restrict__ agg) {
  int e    = blockIdx.x * (blockDim.x >> 5) + (threadIdx.x >> 5);  // edge id
  int lane = threadIdx.x & 31;
  int s = src[e];
  int d = dst[e];
  const float* vr = v   + (size_t)s * HC + lane;
  float*       ar = agg + (size_t)d * HC + lane;
  const float* av = alpha + (size_t)e * NHEAD;
#pragma unroll
  for (int h = 0; h < NHEAD; ++h) {
    atomicAdd(&ar[h * 32], vr[h * 32] * av[h]);
  }
}

// out[n,c] = relu( mean_h(agg[n,h,c]) + skip[n,c] )
__global__ void finalize(const float* __restrict__ agg, const float* __restrict__ skip,
                         float* __restrict__ out) {
  int i = blockIdx.x * blockDim.x + threadIdx.x;   // n*32 + c
  if (i >= NNODES * HIDC) return;
  int n = i >> 5;
  int c = i & 31;
  const float* ar = agg + (size_t)n * HC + c;
  float s = (ar[0] + ar[32] + ar[64] + ar[96]) * 0.25f + skip[i];
  out[i] = s > 0.0f ? s : 0.0f;
}

// g[b,c] = mean over NODE_PER nodes of h[b*NODE_PER + n][c]
__global__ void readout_mean(const float* __restrict__ h, float* __restrict__ g) {
  int b = blockIdx.x / EMBC;
  int c = blockIdx.x % EMBC;
  float s = 0.0f;
  for (int n = threadIdx.x; n < NODE_PER; n += blockDim.x)
    s += h[((size_t)b * NODE_PER + n) * EMBC + c];
  __shared__ float red[256];
  red[threadIdx.x] = s;
  __syncthreads();
  for (int off = 128; off > 0; off >>= 1) {
    if (threadIdx.x < off) red[threadIdx.x] += red[threadIdx.x + off];
    __syncthreads();
  }
  if (threadIdx.x == 0) g[blockIdx.x] = red[0] * (1.0f / NODE_PER);
}

// out[b,j] = g[b,:] @ Wout[:,j] + bout[j]
__global__ void out_gemm(const float* __restrict__ g, const float* __restrict__ Wout,
                         const float* __restrict__ bout, float* __restrict__ out) {
  int i = threadIdx.x;
  if (i < NB * NCLS) {
    int b = i / NCLS, j = i % NCLS;
    float s = bout[j];
    for (int c = 0; c < EMBC; ++c) s += g[b * EMBC + c] * Wout[c * NCLS + j];
    out[i] = s;
  }
}

// ---------------------------------------------------------------------------
extern "C" void kernel_launch(void* const* d_in, const int* in_sizes, int n_in,
                              void* d_out, int out_size, void* d_ws, size_t ws_size,
                              hipStream_t stream) {
  (void)in_sizes; (void)n_in; (void)out_size; (void)ws_size;

  const float* x  = (const float*)d_in[0];
  const int*   ei = (const int*)d_in[1];
  const int*   src = ei;
  const int*   dst = ei + NEDGES;

  const float *Wq[3], *bq[3], *Wk[3], *bk[3], *Wv[3], *bv[3], *Ws[3], *bs[3];
  for (int L = 0; L < 3; ++L) {
    int base = 2 + 8 * L;
    Wq[L] = (const float*)d_in[base + 0];
    bq[L] = (const float*)d_in[base + 1];
    Wk[L] = (const float*)d_in[base + 2];
    bk[L] = (const float*)d_in[base + 3];
    Wv[L] = (const float*)d_in[base + 4];
    bv[L] = (const float*)d_in[base + 5];
    Ws[L] = (const float*)d_in[base + 6];
    bs[L] = (const float*)d_in[base + 7];
  }
  const float* Wout = (const float*)d_in[26];
  const float* bout = (const float*)d_in[27];

  // Workspace carve-out (256B aligned)
  char* w = (char*)d_ws;
  auto carve = [&](size_t bytes) {
    char* p = w;
    w += (bytes + 255) & ~(size_t)255;
    return p;
  };
  float*    q     = (float*)   carve((size_t)NNODES * HC * 4);
  float*    kbuf  = (float*)   carve((size_t)NNODES * HC * 4);
  float*    vbuf  = (float*)   carve((size_t)NNODES * HC * 4);
  float*    skip  = (float*)   carve((size_t)NNODES * HIDC * 4);
  float*    alpha = (float*)   carve((size_t)NEDGES * NHEAD * 4);
  unsigned* mkey  = (unsigned*)carve((size_t)NNODES * NHEAD * 4);
  float*    ssum  = (float*)   carve((size_t)NNODES * NHEAD * 4);
  float*    hA    = (float*)   carve((size_t)NNODES * HIDC * 4);
  float*    hB    = (float*)   carve((size_t)NNODES * HIDC * 4);
  float*    g     = (float*)   carve((size_t)NB * EMBC * 4);
  float*    agg   = q;   // q is dead after edge_alpha; reuse as aggregation buffer

  const int EH         = NEDGES * NHEAD;
  const int edgeBlks   = (EH + 255) / 256;
  const int segBlks    = (NNODES * NHEAD + 255) / 256;
  const int aggBlks    = (NNODES * HC + 255) / 256;
  const int finBlks    = (NNODES * HIDC + 255) / 256;
  const int scatBlks   = NEDGES / 8;                 // wave-per-edge, 8 waves/block
  const int rowTiles   = NNODES / 16;                // 4096

  const float* in = x;
  int K = DIN;
  float* outs[3] = { hA, hB, hA };

  for (int L = 0; L < 3; ++L) {
    // Projections on the WMMA pipe (fp32, exact)
    int blkQKV = rowTiles * (HC / 16) / 4;           // 4 waves / block of 128
    int blkS   = rowTiles * (HIDC / 16) / 4;
    gemm_bias_wmma<<<blkQKV, 128, 0, stream>>>(in, Wq[L], bq[L], q,    K, HC);
    gemm_bias_wmma<<<blkQKV, 128, 0, stream>>>(in, Wk[L], bk[L], kbuf, K, HC);
    gemm_bias_wmma<<<blkQKV, 128, 0, stream>>>(in, Wv[L], bv[L], vbuf, K, HC);
    gemm_bias_wmma<<<blkS,   128, 0, stream>>>(in, Ws[L], bs[L], skip, K, HIDC);

    // Segment softmax + weighted scatter
    init_seg    <<<segBlks,  256, 0, stream>>>(mkey, ssum, NNODES * NHEAD);
    edge_alpha  <<<edgeBlks, 256, 0, stream>>>(q, kbuf, src, dst, alpha);
    zero_f32    <<<aggBlks,  256, 0, stream>>>(agg, NNODES * HC);   // after q consumed
    edge_max    <<<edgeBlks, 256, 0, stream>>>(alpha, dst, mkey);
    edge_expsum <<<edgeBlks, 256, 0, stream>>>(alpha, dst, mkey, ssum);
    edge_norm   <<<edgeBlks, 256, 0, stream>>>(alpha, dst, ssum);
    edge_scatter<<<scatBlks, 256, 0, stream>>>(alpha, src, dst, vbuf, agg);
    finalize    <<<finBlks,  256, 0, stream>>>(agg, skip, outs[L]);

    in = outs[L];
    K = HIDC;
  }

  // Per-graph readout + classifier head
  readout_mean<<<NB * EMBC, 256, 0, stream>>>(outs[2], g);
  out_gemm<<<1, 32, 0, stream>>>(g, Wout, bout, (float*)d_out);
}